// GATWithPositionalEncoding_25769804021
// MI455X (gfx1250) — compile-verified
//
#include <hip/hip_runtime.h>
#include <math.h>

typedef float v2f __attribute__((ext_vector_type(2)));
typedef float v8f __attribute__((ext_vector_type(8)));

namespace {
constexpr int kNodes = 50000;
constexpr int kEdges = 800000;
constexpr int kInCh  = 64;
constexpr int kOutCh = 32;
constexpr int kHeads = 4;
constexpr int kHC    = kHeads * kOutCh;   // 128
constexpr int kETot  = kEdges + kNodes;   // edges + self loops
}

// ---------- helpers ----------

__device__ __forceinline__ void atomic_max_f32(float* addr, float v) {
  // int ordering == float ordering for non-negative floats;
  // reversed unsigned ordering for negative floats.
  if (v >= 0.0f) atomicMax(reinterpret_cast<int*>(addr), __float_as_int(v));
  else           atomicMin(reinterpret_cast<unsigned int*>(addr), __float_as_uint(v));
}

__device__ __forceinline__ float leaky02(float v) { return v > 0.0f ? v : 0.2f * v; }

// ---------- kernels ----------

// x_pos = x
__global__ void k_copy_x(const float* __restrict__ x, float* __restrict__ x_pos) {
  int i = blockIdx.x * blockDim.x + threadIdx.x;
  if (i < kNodes * kInCh) x_pos[i] = x[i];
}

// amax = -inf, denom = 0, out = 0
__global__ void k_init(float* __restrict__ amax, float* __restrict__ denom,
                       float* __restrict__ out) {
  int i = blockIdx.x * blockDim.x + threadIdx.x;
  if (i < kNodes * kHeads) { amax[i] = -INFINITY; denom[i] = 0.0f; }
  if (i < kNodes * kHC)    out[i] = 0.0f;
}

// x_pos[src[e]] += pe[edge_type[e]]   (per edge, per channel)
__global__ void k_scatter_pe(const int* __restrict__ ei, const int* __restrict__ et,
                             const float* __restrict__ pe, float* __restrict__ x_pos) {
  long long tid = (long long)blockIdx.x * blockDim.x + threadIdx.x;
  if (tid >= (long long)kEdges * kInCh) return;
  int e = (int)(tid >> 6);      // /64
  int c = (int)(tid & 63);      // %64
  int s = ei[e];                // src row of edge_index
  atomicAdd(&x_pos[(size_t)s * kInCh + c], pe[(size_t)et[e] * kInCh + c]);
}

// xl = x_pos @ W  via V_WMMA_F32_16X16X4_F32; one wave per 16x16 tile.
__global__ void __launch_bounds__(256) k_gemm_wmma(const float* __restrict__ A,  // [50000,64]
                                                   const float* __restrict__ B,  // [64,128]
                                                   float* __restrict__ C) {      // [50000,128]
  const int lane = threadIdx.x & 31;
  const int wave = threadIdx.x >> 5;     // 0..7 -> N tile
  const int m0   = blockIdx.x * 16;      // 3125 blocks, exact
  const int n0   = wave * 16;
  const int half = lane >> 4;            // 0 | 1
  const int l15  = lane & 15;
  const int row  = m0 + l15;             // A row for this lane
  const int col  = n0 + l15;             // B/C col for this lane

  v8f acc = {0.f, 0.f, 0.f, 0.f, 0.f, 0.f, 0.f, 0.f};
#pragma unroll
  for (int k0 = 0; k0 < kInCh; k0 += 4) {
    const int kb = k0 + 2 * half;        // lanes 0-15: k0,k0+1 ; lanes 16-31: k0+2,k0+3
    v2f a, b;
    a.x = A[(size_t)row * kInCh + kb];
    a.y = A[(size_t)row * kInCh + kb + 1];
    b.x = B[(size_t)kb * kHC + col];
    b.y = B[(size_t)(kb + 1) * kHC + col];
    acc = __builtin_amdgcn_wmma_f32_16x16x4_f32(false, a, false, b, (short)0, acc,
                                                false, false);
  }
#pragma unroll
  for (int r = 0; r < 8; ++r) {
    C[(size_t)(m0 + r + 8 * half) * kHC + col] = acc[r];
  }
}

// per-(node,head) attention scores: a_src[n,h] = sum_c xl[n,h,c]*att_src[h,c]
__global__ void k_scores(const float* __restrict__ xl,
                         const float* __restrict__ att_src, const float* __restrict__ att_dst,
                         float* __restrict__ a_src, float* __restrict__ a_dst) {
  int tid = blockIdx.x * blockDim.x + threadIdx.x;
  if (tid >= kNodes * kHeads) return;
  int n = tid / kHeads, h = tid % kHeads;
  const float* xr = xl + (size_t)n * kHC + h * kOutCh;
  float ss = 0.0f, sd = 0.0f;
#pragma unroll
  for (int c = 0; c < kOutCh; ++c) {
    ss += xr[c] * att_src[h * kOutCh + c];
    sd += xr[c] * att_dst[h * kOutCh + c];
  }
  a_src[tid] = ss;
  a_dst[tid] = sd;
}

__device__ __forceinline__ void edge_sd(const int* __restrict__ ei, int e, int& s, int& d) {
  if (e < kEdges) { s = ei[e]; d = ei[kEdges + e]; }
  else            { s = e - kEdges; d = s; }       // self loop
}

// pass 1: segment max of leaky_relu(a_src[s]+a_dst[d]) over destinations
__global__ void k_alpha_max(const int* __restrict__ ei,
                            const float* __restrict__ a_src, const float* __restrict__ a_dst,
                            float* __restrict__ amax) {
  long long tid = (long long)blockIdx.x * blockDim.x + threadIdx.x;
  if (tid >= (long long)kETot * kHeads) return;
  int e = (int)(tid >> 2), h = (int)(tid & 3);
  int s, d; edge_sd(ei, e, s, d);
  float al = leaky02(a_src[s * kHeads + h] + a_dst[d * kHeads + h]);
  atomic_max_f32(&amax[d * kHeads + h], al);
}

// pass 2: denom[d,h] += exp(alpha - amax[d,h])
__global__ void k_exp_sum(const int* __restrict__ ei,
                          const float* __restrict__ a_src, const float* __restrict__ a_dst,
                          const float* __restrict__ amax, float* __restrict__ denom) {
  long long tid = (long long)blockIdx.x * blockDim.x + threadIdx.x;
  if (tid >= (long long)kETot * kHeads) return;
  int e = (int)(tid >> 2), h = (int)(tid & 3);
  int s, d; edge_sd(ei, e, s, d);
  float al = leaky02(a_src[s * kHeads + h] + a_dst[d * kHeads + h]);
  atomicAdd(&denom[d * kHeads + h], expf(al - amax[d * kHeads + h]));
}

// pass 3: out[d,h,:] += coef * xl[s,h,:]
__global__ void k_aggregate(const int* __restrict__ ei,
                            const float* __restrict__ a_src, const float* __restrict__ a_dst,
                            const float* __restrict__ amax, const float* __restrict__ denom,
                            const float* __restrict__ xl, float* __restrict__ out) {
  long long tid = (long long)blockIdx.x * blockDim.x + threadIdx.x;
  if (tid >= (long long)kETot * kHeads) return;
  int e = (int)(tid >> 2), h = (int)(tid & 3);
  int s, d; edge_sd(ei, e, s, d);
  float al = leaky02(a_src[s * kHeads + h] + a_dst[d * kHeads + h]);
  float coef = expf(al - amax[d * kHeads + h]) / denom[d * kHeads + h];
  const float* xr = xl + (size_t)s * kHC + h * kOutCh;
  float* orow = out + (size_t)d * kHC + h * kOutCh;
#pragma unroll 8
  for (int c = 0; c < kOutCh; ++c) atomicAdd(&orow[c], coef * xr[c]);
}

// out = elu(out + bias)
__global__ void k_finalize(float* __restrict__ out, const float* __restrict__ bias) {
  int i = blockIdx.x * blockDim.x + threadIdx.x;
  if (i >= kNodes * kHC) return;
  float v = out[i] + bias[i & (kHC - 1)];
  out[i] = v > 0.0f ? v : expm1f(v);
}

// ---------- launch ----------

extern "C" void kernel_launch(void* const* d_in, const int* in_sizes, int n_in,
                              void* d_out, int out_size, void* d_ws, size_t ws_size,
                              hipStream_t stream) {
  (void)in_sizes; (void)n_in; (void)out_size; (void)ws_size;
  const float* x       = (const float*)d_in[0];
  const int*   ei      = (const int*)d_in[1];   // [2, E]: row0=src, row1=dst
  const int*   et      = (const int*)d_in[2];   // [E]
  const float* pe      = (const float*)d_in[3]; // [64, 64]
  const float* W       = (const float*)d_in[4]; // [64, 128]
  const float* att_src = (const float*)d_in[5]; // [4, 32]
  const float* att_dst = (const float*)d_in[6]; // [4, 32]
  const float* bias    = (const float*)d_in[7]; // [128]
  float* out = (float*)d_out;                   // [50000, 128]

  float* ws    = (float*)d_ws;
  float* x_pos = ws;                                      // 3.2e6 f
  float* xl    = x_pos + (size_t)kNodes * kInCh;          // 6.4e6 f
  float* a_src = xl    + (size_t)kNodes * kHC;            // 2.0e5 f
  float* a_dst = a_src + (size_t)kNodes * kHeads;
  float* amax  = a_dst + (size_t)kNodes * kHeads;
  float* denom = amax  + (size_t)kNodes * kHeads;

  const int B = 256;
  k_copy_x<<<(kNodes * kInCh + B - 1) / B, B, 0, stream>>>(x, x_pos);
  k_init<<<(kNodes * kHC + B - 1) / B, B, 0, stream>>>(amax, denom, out);
  {
    long long n = (long long)kEdges * kInCh;
    k_scatter_pe<<<(unsigned)((n + B - 1) / B), B, 0, stream>>>(ei, et, pe, x_pos);
  }
  k_gemm_wmma<<<kNodes / 16, 256, 0, stream>>>(x_pos, W, xl);
  k_scores<<<(kNodes * kHeads + B - 1) / B, B, 0, stream>>>(xl, att_src, att_dst, a_src, a_dst);
  {
    long long n = (long long)kETot * kHeads;
    unsigned g = (unsigned)((n + B - 1) / B);
    k_alpha_max<<<g, B, 0, stream>>>(ei, a_src, a_dst, amax);
    k_exp_sum<<<g, B, 0, stream>>>(ei, a_src, a_dst, amax, denom);
    k_aggregate<<<g, B, 0, stream>>>(ei, a_src, a_dst, amax, denom, xl, out);
  }
  k_finalize<<<(kNodes * kHC + B - 1) / B, B, 0, stream>>>(out, bias);
}